// WeightGenerator_V4_807453852205
// MI455X (gfx1250) — compile-verified
//
#include <hip/hip_runtime.h>
#include <hip/hip_bf16.h>
#include <stddef.h>

typedef __attribute__((ext_vector_type(2))) float v2f;
typedef __attribute__((ext_vector_type(8))) float v8f;

// ---------------------------------------------------------------------------
// Kernel A: channel_reduce grouped conv 3x3 (256 -> 64, groups=64, pad=1)
// grid = B*64 (one block per (b, oc)), block = 256 threads (one per pixel)
// ---------------------------------------------------------------------------
__global__ void wg_xr_kernel(const float* __restrict__ x,
                             const float* __restrict__ w_cr,
                             const float* __restrict__ b_cr,
                             float* __restrict__ xr) {
  const int bc = blockIdx.x;          // b*64 + oc
  const int b  = bc >> 6;
  const int oc = bc & 63;
  const int t  = threadIdx.x;         // pixel 0..255
  const int h  = t >> 4, w = t & 15;
  float s = b_cr[oc];
  for (int k = 0; k < 4; ++k) {
    const float* xp = x + (size_t)(b * 256 + oc * 4 + k) * 256;
    const float* wp = w_cr + (size_t)(oc * 4 + k) * 9;
#pragma unroll
    for (int dy = 0; dy < 3; ++dy) {
      const int hh = h + dy - 1;
      if (hh < 0 || hh > 15) continue;
#pragma unroll
      for (int dx = 0; dx < 3; ++dx) {
        const int ww = w + dx - 1;
        if (ww < 0 || ww > 15) continue;
        s += wp[dy * 3 + dx] * xp[hh * 16 + ww];
      }
    }
  }
  xr[(size_t)bc * 256 + t] = s;
}

// ---------------------------------------------------------------------------
// Kernel B: gate (1x1 conv + BN + StarReLU) fused with depthwise 3x3 / 5x5
// and the gating multiply -> xc[b, pix, ch] (ch-contiguous for kernel C)
// grid = B*256 (one block per (b, pixel)), block = 128 threads (one per ch)
// ---------------------------------------------------------------------------
__global__ void wg_xc_kernel(const float* __restrict__ x,
                             const float* __restrict__ xr,
                             const float* __restrict__ w_dw3, const float* __restrict__ b_dw3,
                             const float* __restrict__ w_dw7, const float* __restrict__ b_dw7,
                             const float* __restrict__ w_gate, const float* __restrict__ b_gate,
                             const float* __restrict__ bn_gamma, const float* __restrict__ bn_beta,
                             const float* __restrict__ bn_mean, const float* __restrict__ bn_var,
                             const float* __restrict__ sr_scale, const float* __restrict__ sr_bias,
                             float* __restrict__ xcbuf) {
  __shared__ float xs[256];
  const int bp  = blockIdx.x;         // b*256 + pixel
  const int b   = bp >> 8;
  const int pix = bp & 255;
  const int h   = pix >> 4, w = pix & 15;
  const int t   = threadIdx.x;        // 0..127
  xs[t]       = x[(size_t)(b * 256 + t) * 256 + pix];
  xs[t + 128] = x[(size_t)(b * 256 + t + 128) * 256 + pix];
  __syncthreads();

  // gate branch
  float a = b_gate[t];
  const float* wg = w_gate + (size_t)t * 256;
  for (int ic = 0; ic < 256; ++ic) a += wg[ic] * xs[ic];
  const float inv = bn_gamma[t] * rsqrtf(bn_var[t] + 1e-5f);
  float gv = (a - bn_mean[t]) * inv + bn_beta[t];
  gv = fmaxf(gv, 0.0f);
  gv = sr_scale[0] * gv * gv + sr_bias[0];

  // feature branch: depthwise conv on xr channel (t % 64)
  const int ch = t & 63;
  const float* xp = xr + (size_t)(b * 64 + ch) * 256;
  float s;
  if (t < 64) {
    s = b_dw3[ch];
    const float* wp = w_dw3 + ch * 9;
#pragma unroll
    for (int dy = 0; dy < 3; ++dy) {
      const int hh = h + dy - 1;
      if (hh < 0 || hh > 15) continue;
#pragma unroll
      for (int dx = 0; dx < 3; ++dx) {
        const int ww = w + dx - 1;
        if (ww < 0 || ww > 15) continue;
        s += wp[dy * 3 + dx] * xp[hh * 16 + ww];
      }
    }
  } else {
    s = b_dw7[ch];
    const float* wp = w_dw7 + ch * 25;
#pragma unroll
    for (int dy = 0; dy < 5; ++dy) {
      const int hh = h + dy - 2;
      if (hh < 0 || hh > 15) continue;
#pragma unroll
      for (int dx = 0; dx < 5; ++dx) {
        const int ww = w + dx - 2;
        if (ww < 0 || ww > 15) continue;
        s += wp[dy * 5 + dx] * xp[hh * 16 + ww];
      }
    }
  }
  xcbuf[(size_t)bp * 128 + t] = s * gv;
}

// ---------------------------------------------------------------------------
// Kernel C: per-patch  out(256e x 64p) = wdec(256e x 256c) @ y_patch(256c x 64p)
// with wdec generated ON THE MATRIX ENGINE, chunked through LDS:
//
//  phase 1 (per 64-c chunk): for each oh (32):
//     wdec_oh[g, c] = XC(8x16,pad16) @ W_oh(16x64c) + b_lin   (f32 WMMA)
//     stored to LDS Wd2[e' = oh*8+g][cc], row stride 68 floats
//  phase 2: main matmul; A operand = single conflict-free ds_load_b64 from Wd2
//     (stride 68 == 4 mod 64  -> both half-waves on disjoint banks),
//     B operand = ds_load_b64 from c-pair-interleaved Ys (stride 160 == 32 mod 64).
//
// M-tile row permutation: tile t covers e' in [16t,16t+16)  <->  oh {2t,2t+1},
// all 8 groups, so weight-gen output order matches main-matmul A order.
// ---------------------------------------------------------------------------
#define YS_STRIDE 160
#define WD_STRIDE 68
#define YS_FLOATS (128 * YS_STRIDE)          // 20480
#define WD_FLOATS (256 * WD_STRIDE)          // 17408

__launch_bounds__(256, 2)
__global__ void wg_main_kernel(const float* __restrict__ y,
                               const float* __restrict__ xcbuf,
                               const float* __restrict__ w_lin,
                               const float* __restrict__ b_lin,
                               float* __restrict__ out) {
  extern __shared__ float smem[];
  float* Ys  = smem;                       // 80 KB
  float* Wd2 = smem + YS_FLOATS;           // 68 KB
  float* Xcs = smem + YS_FLOATS + WD_FLOATS;  // 128 floats

  const int tid   = threadIdx.x;
  const int b     = blockIdx.x >> 8;
  const int patch = blockIdx.x & 255;
  const int f  = patch >> 4;
  const int gc = patch & 15;

  if (tid < 128) Xcs[tid] = xcbuf[(size_t)(b * 256 + patch) * 128 + tid];

  // stage y patch: thread = (p = tid&63, cg = tid>>6), channels [64cg, 64cg+64)
  {
    const int p  = tid & 63, cg = tid >> 6;
    const int pr = p >> 3,  pc = p & 7;
    const float* yb = y + (size_t)b * 256 * 16384 + (size_t)(f * 8 + pr) * 128 + gc * 8 + pc;
    for (int cc = 0; cc < 64; ++cc) {
      const int c = cg * 64 + cc;
      Ys[(c >> 1) * YS_STRIDE + 2 * p + (c & 1)] = yb[(size_t)c * 16384];
    }
  }
  __syncthreads();

  const int lane = tid & 31;
  const int wv   = tid >> 5;      // wave 0..7
  const int m    = lane & 15;
  const int hi   = lane >> 4;     // half-wave
  const int koff = hi << 1;       // K offset 0/2 (f32 WMMA operand layout)

  // weight-gen A operand: XC tile (rows m = g, rows >= 8 are zero), 4 K-steps
  v2f a_wg[4];
#pragma unroll
  for (int kk = 0; kk < 4; ++kk) {
    float ax = 0.0f, ay = 0.0f;
    if (m < 8) {
      ax = Xcs[m * 16 + kk * 4 + koff];
      ay = Xcs[m * 16 + kk * 4 + koff + 1];
    }
    a_wg[kk].x = ax; a_wg[kk].y = ay;
  }

  v8f acc[2][4];
#pragma unroll
  for (int mt = 0; mt < 2; ++mt)
#pragma unroll
    for (int nt = 0; nt < 4; ++nt)
      acc[mt][nt] = (v8f)0.0f;

  const int ebase0 = 16 * wv;         // M-tile t = wv     (e' rows)
  const int ebase1 = 16 * wv + 128;   // M-tile t = wv + 8
  const int ohbase0 = 2 * wv;
  const int ohbase1 = 2 * wv + 16;

  for (int chunk = 0; chunk < 4; ++chunk) {
    const int cbase = chunk * 64;

    // ---- phase 1: weight generation into Wd2 (wave wv owns oh = 4wv..4wv+3)
#pragma unroll
    for (int ohi = 0; ohi < 4; ++ohi) {
      const int oh = 4 * wv + ohi;
#pragma unroll
      for (int ct = 0; ct < 4; ++ct) {
        const int c0 = cbase + ct * 16;
        const float bl = b_lin[oh * 256 + c0 + m];
        v8f d;
#pragma unroll
        for (int j = 0; j < 8; ++j) d[j] = bl;
        const float* wb = w_lin + (size_t)(oh * 256 + c0 + m) * 16 + koff;
#pragma unroll
        for (int kk = 0; kk < 4; ++kk) {
          const v2f Bw = *(const v2f*)(wb + kk * 4);
          d = __builtin_amdgcn_wmma_f32_16x16x4_f32(
              false, a_wg[kk], false, Bw, (short)0, d, false, false);
        }
        if (lane < 16) {   // rows g = 0..7 live in lanes 0..15 of VGPRs 0..7
          float* wrow = &Wd2[(oh * 8) * WD_STRIDE + (c0 - cbase) + m];
#pragma unroll
          for (int j = 0; j < 8; ++j) wrow[j * WD_STRIDE] = d[j];
        }
      }
    }
    __syncthreads();

    // ---- phase 2: main matmul over this 64-c chunk
    for (int cq = 0; cq < 64; cq += 4) {
      const int cb = cbase + cq + koff;     // global c (even)
      const int cp = cb >> 1;
      v2f Bt[4];
#pragma unroll
      for (int nt = 0; nt < 4; ++nt) {
        const int pp = nt * 16 + m;
        Bt[nt] = *(const v2f*)&Ys[cp * YS_STRIDE + 2 * pp];
      }
      const v2f A0 = *(const v2f*)&Wd2[(ebase0 + m) * WD_STRIDE + cq + koff];
      const v2f A1 = *(const v2f*)&Wd2[(ebase1 + m) * WD_STRIDE + cq + koff];
#pragma unroll
      for (int nt = 0; nt < 4; ++nt) {
        acc[0][nt] = __builtin_amdgcn_wmma_f32_16x16x4_f32(
            false, A0, false, Bt[nt], (short)0, acc[0][nt], false, false);
        acc[1][nt] = __builtin_amdgcn_wmma_f32_16x16x4_f32(
            false, A1, false, Bt[nt], (short)0, acc[1][nt], false, false);
      }
    }
    __syncthreads();   // before next chunk overwrites Wd2
  }

  // store: C layout -> VGPR j holds M = j (lanes 0-15) / j+8 (lanes 16-31)
#pragma unroll
  for (int mt = 0; mt < 2; ++mt) {
    const int ohb = mt ? ohbase1 : ohbase0;
#pragma unroll
    for (int nt = 0; nt < 4; ++nt) {
#pragma unroll
      for (int j = 0; j < 8; ++j) {
        const int mm  = j + (hi << 3);
        const int e   = (mm & 7) * 32 + ohb + (mm >> 3);  // undo row permutation
        const int pp  = nt * 16 + m;
        const int prr = pp >> 3, pcc = pp & 7;
        out[((size_t)(b * 256 + e) * 128 + f * 8 + prr) * 128 + gc * 8 + pcc] =
            acc[mt][nt][j];
      }
    }
  }
}

// ---------------------------------------------------------------------------
extern "C" void kernel_launch(void* const* d_in, const int* in_sizes, int n_in,
                              void* d_out, int out_size, void* d_ws, size_t ws_size,
                              hipStream_t stream) {
  const float* x        = (const float*)d_in[0];
  const float* y        = (const float*)d_in[1];
  const float* w_cr     = (const float*)d_in[2];
  const float* b_cr     = (const float*)d_in[3];
  const float* w_dw3    = (const float*)d_in[4];
  const float* b_dw3    = (const float*)d_in[5];
  const float* w_dw7    = (const float*)d_in[6];
  const float* b_dw7    = (const float*)d_in[7];
  const float* w_gate   = (const float*)d_in[8];
  const float* b_gate   = (const float*)d_in[9];
  const float* bn_gamma = (const float*)d_in[10];
  const float* bn_beta  = (const float*)d_in[11];
  const float* bn_mean  = (const float*)d_in[12];
  const float* bn_var   = (const float*)d_in[13];
  const float* sr_scale = (const float*)d_in[14];
  const float* sr_bias  = (const float*)d_in[15];
  const float* w_lin    = (const float*)d_in[16];
  const float* b_lin    = (const float*)d_in[17];
  float* out = (float*)d_out;

  float* xr  = (float*)d_ws;          // 2*64*256   = 32768 floats
  float* xcb = xr + 2 * 64 * 256;     // 2*256*128  = 65536 floats

  wg_xr_kernel<<<dim3(2 * 64), dim3(256), 0, stream>>>(x, w_cr, b_cr, xr);
  wg_xc_kernel<<<dim3(2 * 256), dim3(128), 0, stream>>>(
      x, xr, w_dw3, b_dw3, w_dw7, b_dw7, w_gate, b_gate,
      bn_gamma, bn_beta, bn_mean, bn_var, sr_scale, sr_bias, xcb);

  const size_t lds_bytes = (size_t)(YS_FLOATS + WD_FLOATS + 128) * sizeof(float);
  wg_main_kernel<<<dim3(2 * 256), dim3(256), lds_bytes, stream>>>(
      y, xcb, w_lin, b_lin, out);
}